// DWT_2D_41059887350275
// MI455X (gfx1250) — compile-verified
//
#include <hip/hip_runtime.h>

typedef float v2f __attribute__((ext_vector_type(2)));
typedef float v8f __attribute__((ext_vector_type(8)));

// Problem constants (fixed by the reference setup)
constexpr int Bsz = 8, C = 256, H = 224, W = 224;
constexpr int G   = C / 4;           // 64 groups
constexpr int H2  = H / 2;           // 112
constexpr int W2  = W / 2;           // 112
constexpr int JT  = W2 / 16;         // 7 tiles of 16 output columns

// One wave32 per output row (b, g, i).
//   Lanes  0-15 : pixel n = lane, reads input row 2i  -> B rows K0 (a), K1 (b)
//   Lanes 16-31 : pixel n = lane-16, reads input row 2i+1 -> B rows K2 (c), K3 (d)
// A (16x4 f32): rows 0..3 = weight rows of (ll, lh, hl, hh); rows 4..15 zero.
//   A layout: lanes 0-15 hold K=0,1 in v0,v1; lanes 16-31 hold K=2,3.
// D (16x16 f32): VGPR r, lanes 0-15 => M=r, N=lane. We use M=0..3 -> the four
// output planes for 16 pixels, with no cross-lane data movement anywhere.
__global__ __launch_bounds__(256)
void haar_dwt_wmma_kernel(const float* __restrict__ x,
                          const float* __restrict__ wll,
                          const float* __restrict__ wlh,
                          const float* __restrict__ whl,
                          const float* __restrict__ whh,
                          float* __restrict__ out)
{
    const int lane      = threadIdx.x & 31;
    const int waveInBlk = threadIdx.x >> 5;
    const int wave      = blockIdx.x * 8 + waveInBlk;   // 0 .. B*G*H2-1 exactly

    const int i  = wave % H2;
    const int bg = wave / H2;
    const int g  = bg % G;
    const int b  = bg / G;

    const int m  = lane & 15;   // pixel within tile / A-matrix row
    const int kh = lane >> 4;   // 0 -> K{0,1}=(a,b);  1 -> K{2,3}=(c,d)

    // ---- A matrix: per-lane two weight values (zero for rows 4..15)
    v2f A = {0.0f, 0.0f};
    {
        const float* wsel = (m == 0) ? wll : (m == 1) ? wlh : (m == 2) ? whl : whh;
        if (m < 4) {
            A.x = wsel[2 * kh + 0];   // weight for K = 2*kh   (a or c)
            A.y = wsel[2 * kh + 1];   // weight for K = 2*kh+1 (b or d)
        }
    }

    // Input base: this lane's row (2i + kh) of channel 4g, batch b
    const int row = 2 * i + kh;
    const float* xbase = x + (((size_t)(b * C + 4 * g) * H + row) * W);
    constexpr size_t chStride = (size_t)H * W;

    // Output base: channel plane g of type 0; types are 64 planes apart
    float* obase = out + (((size_t)(b * C + g) * H2 + i) * W2);
    constexpr size_t tStride = (size_t)G * H2 * W2;

    for (int jt = 0; jt < JT; ++jt) {
        const int col = jt * 32 + 2 * m;   // input column of this lane's pixel

        // Prefetch next tile of channel 0 (gfx1250 global_prefetch_b8)
        if (jt + 1 < JT)
            __builtin_prefetch(xbase + col + 32, 0, 1);

        // Channel-group sum: lane<16 -> (Σa, Σb), lane>=16 -> (Σc, Σd) == B column
        v2f s = {0.0f, 0.0f};
        #pragma unroll
        for (int ch = 0; ch < 4; ++ch) {
            const float2 v = *reinterpret_cast<const float2*>(xbase + ch * chStride + col);
            s.x += v.x;
            s.y += v.y;
        }

        // D(16x16) = A(16x4) x B(4x16): rows 0..3 = ll,lh,hl,hh for 16 pixels
        v8f acc = {};
        acc = __builtin_amdgcn_wmma_f32_16x16x4_f32(
            /*neg_a=*/false, A, /*neg_b=*/false, s,
            /*c_mod=*/(short)0, acc, /*reuse_a=*/false, /*reuse_b=*/false);

        // Lanes 0-15 hold D rows 0..3 for pixel n = lane: 4 contiguous 64B stores
        if (lane < 16) {
            const int j = jt * 16 + m;
            obase[0 * tStride + j] = acc[0];
            obase[1 * tStride + j] = acc[1];
            obase[2 * tStride + j] = acc[2];
            obase[3 * tStride + j] = acc[3];
        }
    }
}

extern "C" void kernel_launch(void* const* d_in, const int* in_sizes, int n_in,
                              void* d_out, int out_size, void* d_ws, size_t ws_size,
                              hipStream_t stream) {
    const float* x   = (const float*)d_in[0];
    const float* wll = (const float*)d_in[1];
    const float* wlh = (const float*)d_in[2];
    const float* whl = (const float*)d_in[3];
    const float* whh = (const float*)d_in[4];
    float* out = (float*)d_out;

    const int totalWaves = Bsz * G * H2;       // 57344, divisible by 8
    const int blocks     = totalWaves / 8;     // 7168 blocks of 8 waves
    hipLaunchKernelGGL(haar_dwt_wmma_kernel, dim3(blocks), dim3(256), 0, stream,
                       x, wll, wlh, whl, whh, out);
}